// Model9_64630667870278
// MI455X (gfx1250) — compile-verified
//
#include <hip/hip_runtime.h>

typedef __attribute__((ext_vector_type(2))) float v2f;
typedef __attribute__((ext_vector_type(8))) float v8f;

#define N_NODES  50000
#define N_EDGES  800000
#define D_FEAT   64
#define N_GRAPHS 512
#define H1 20
#define H2 10

// ---------------------------------------------------------------------------
// Kernel 1: edge MLP (concat[192] @ W_mpl[192,20] + b, ReLU) + scatter-add.
// 256 threads = 8 waves; each wave computes a 32-edge x 32-col tile with
// V_WMMA_F32_16X16X4_F32: per K-step (K+=4) -> 2 A b64 loads, 2 B ds_b64
// loads (pre-swizzled layout), 4 WMMAs (2 M-tiles x 2 N-tiles).
// ---------------------------------------------------------------------------
__global__ __launch_bounds__(256)
void edge_mpl_kernel(const int*   __restrict__ edge_index,
                     const float* __restrict__ node_attr,
                     const float* __restrict__ edge_attr,
                     const float* __restrict__ W_mpl,
                     const float* __restrict__ b_mpl,
                     float*       __restrict__ x_nodes)
{
    // Pre-swizzled weights: Wpack[(k4*2 + h)*64 + n*2 + j] = W[(4*k4 + h + 2*j), n]
    // so lane (half,m) reads its B fragment for K-step k = 4*k4, N-tile t as one
    // contiguous b64 at ((k4*2 + half)*32 + t*16 + m)*2.
    __shared__ float Wpack[48 * 2 * 32 * 2];   // 24 KB
    __shared__ float blds[32];
    __shared__ int   dlds[8 * 32];             // dst node per row, per wave

    const int tid = threadIdx.x;

    for (int i = tid; i < 48 * 2 * 32 * 2; i += 256) {
        const int j  = i & 1;
        const int n  = (i >> 1) & 31;
        const int h  = (i >> 6) & 1;
        const int k4 = i >> 7;
        const int k  = k4 * 4 + h + 2 * j;     // source row in W_mpl
        Wpack[i] = (n < H1) ? W_mpl[k * H1 + n] : 0.0f;
    }
    if (tid < 32) blds[tid] = (tid < H1) ? b_mpl[tid] : 0.0f;

    const int wid  = tid >> 5;      // wave in block
    const int lane = tid & 31;
    const int m    = lane & 15;     // row within each 16-edge M-tile
    const int half = lane >> 4;     // lane half selects K pair / M half

    const long long e0  = ((long long)blockIdx.x * 8 + wid) * 32;
    const long long emA = e0 + m;          // M-tile A rows
    const long long emB = e0 + 16 + m;     // M-tile B rows

    const int srcA = edge_index[emA];
    const int dstA = edge_index[(long long)N_EDGES + emA];
    const int srcB = edge_index[emB];
    const int dstB = edge_index[(long long)N_EDGES + emB];
    if (half == 0) {
        dlds[wid * 32 + m]      = dstA;
        dlds[wid * 32 + 16 + m] = dstB;
    }
    __syncthreads();

    v8f accA0 = {};   // tile A, cols 0..15
    v8f accA1 = {};   // tile A, cols 16..31 (16..19 valid)
    v8f accB0 = {};   // tile B, cols 0..15
    v8f accB1 = {};   // tile B, cols 16..31

    const float* baseA[3];
    const float* baseB[3];
    baseA[0] = node_attr + (long long)srcA * D_FEAT;   // K   0..63
    baseA[1] = node_attr + (long long)dstA * D_FEAT;   // K  64..127
    baseA[2] = edge_attr + emA * D_FEAT;               // K 128..191
    baseB[0] = node_attr + (long long)srcB * D_FEAT;
    baseB[1] = node_attr + (long long)dstB * D_FEAT;
    baseB[2] = edge_attr + emB * D_FEAT;

    const float* wp = Wpack + (size_t)half * 64 + (size_t)m * 2;

    for (int seg = 0; seg < 3; ++seg) {
        const float* bA = baseA[seg];
        const float* bB = baseB[seg];
        #pragma unroll
        for (int kk = 0; kk < D_FEAT; kk += 4) {
            const int k4 = (seg * D_FEAT + kk) >> 2;
            // A 16x4 f32: lanes 0-15 hold K = k..k+1, lanes 16-31 K = k+2..k+3
            v2f aA = *(const v2f*)(bA + kk + half * 2);
            v2f aB = *(const v2f*)(bB + kk + half * 2);
            // B 4x16 f32 fragments, one contiguous b64 each
            v2f b0 = *(const v2f*)(wp + (size_t)k4 * 128);
            v2f b1 = *(const v2f*)(wp + (size_t)k4 * 128 + 32);
            accA0 = __builtin_amdgcn_wmma_f32_16x16x4_f32(
                        false, aA, false, b0, (short)0, accA0, false, false);
            accA1 = __builtin_amdgcn_wmma_f32_16x16x4_f32(
                        false, aA, false, b1, (short)0, accA1, false, false);
            accB0 = __builtin_amdgcn_wmma_f32_16x16x4_f32(
                        false, aB, false, b0, (short)0, accB0, false, false);
            accB1 = __builtin_amdgcn_wmma_f32_16x16x4_f32(
                        false, aB, false, b1, (short)0, accB1, false, false);
        }
    }

    // D layout: vgpr r, lane l -> M = r + 8*half, N = l&15 (+16 for *1 accs)
    const float bias0 = blds[m];
    const float bias1 = blds[16 + m];
    #pragma unroll
    for (int r = 0; r < 8; ++r) {
        const int mm = r + 8 * half;
        const int dA = dlds[wid * 32 + mm];
        const int dB = dlds[wid * 32 + 16 + mm];

        float v = accA0[r] + bias0;
        v = v > 0.0f ? v : 0.0f;
        atomicAdd(x_nodes + (long long)dA * H1 + m, v);
        v = accB0[r] + bias0;
        v = v > 0.0f ? v : 0.0f;
        atomicAdd(x_nodes + (long long)dB * H1 + m, v);

        if (m < 4) {   // cols 16..19
            float w = accA1[r] + bias1;
            w = w > 0.0f ? w : 0.0f;
            atomicAdd(x_nodes + (long long)dA * H1 + 16 + m, w);
            w = accB1[r] + bias1;
            w = w > 0.0f ? w : 0.0f;
            atomicAdd(x_nodes + (long long)dB * H1 + 16 + m, w);
        }
    }
}

// ---------------------------------------------------------------------------
// Kernel 2: per-node MLP 20->20->10 (ReLU) + scatter-add into per-graph sums.
// ---------------------------------------------------------------------------
__global__ __launch_bounds__(256)
void node_mlp_kernel(const float* __restrict__ x_nodes,
                     const int*   __restrict__ batch,
                     const float* __restrict__ W1, const float* __restrict__ b1,
                     const float* __restrict__ W2, const float* __restrict__ b2,
                     float*       __restrict__ g)
{
    __shared__ float w1[H1 * H1], bb1[H1], w2[H1 * H2], bb2[H2];
    const int tid = threadIdx.x;
    for (int i = tid; i < H1 * H1; i += 256) w1[i] = W1[i];
    for (int i = tid; i < H1 * H2; i += 256) w2[i] = W2[i];
    if (tid < H1) bb1[tid] = b1[tid];
    if (tid < H2) bb2[tid] = b2[tid];
    __syncthreads();

    const long long i = (long long)blockIdx.x * 256 + tid;
    if (i >= N_NODES) return;

    float x[H1], h[H1];
    #pragma unroll
    for (int k = 0; k < H1; ++k) x[k] = x_nodes[i * H1 + k];

    #pragma unroll
    for (int n = 0; n < H1; ++n) {
        float s = bb1[n];
        #pragma unroll
        for (int k = 0; k < H1; ++k) s += x[k] * w1[k * H1 + n];
        h[n] = s > 0.0f ? s : 0.0f;
    }

    const int gi = batch[i];
    #pragma unroll
    for (int n = 0; n < H2; ++n) {
        float s = bb2[n];
        #pragma unroll
        for (int k = 0; k < H1; ++k) s += h[k] * w2[k * H2 + n];
        s = s > 0.0f ? s : 0.0f;
        atomicAdd(g + (long long)gi * H2 + n, s);
    }
}

// ---------------------------------------------------------------------------
// Kernel 3: per-graph MLP 10->10 (ReLU) -> 1.
// ---------------------------------------------------------------------------
__global__ __launch_bounds__(256)
void graph_mlp_kernel(const float* __restrict__ g,
                      const float* __restrict__ W3, const float* __restrict__ b3,
                      const float* __restrict__ W4, const float* __restrict__ b4,
                      float*       __restrict__ out)
{
    const int i = blockIdx.x * 256 + threadIdx.x;
    if (i >= N_GRAPHS) return;

    float y[H2];
    #pragma unroll
    for (int n = 0; n < H2; ++n) {
        float s = b3[n];
        #pragma unroll
        for (int k = 0; k < H2; ++k) s += g[i * H2 + k] * W3[k * H2 + n];
        y[n] = s > 0.0f ? s : 0.0f;
    }
    float s = b4[0];
    #pragma unroll
    for (int k = 0; k < H2; ++k) s += y[k] * W4[k];
    out[i] = s;
}

// ---------------------------------------------------------------------------
extern "C" void kernel_launch(void* const* d_in, const int* in_sizes, int n_in,
                              void* d_out, int out_size, void* d_ws, size_t ws_size,
                              hipStream_t stream)
{
    const int*   edge_index = (const int*)  d_in[0];
    const float* node_attr  = (const float*)d_in[1];
    const float* edge_attr  = (const float*)d_in[2];
    const int*   batch      = (const int*)  d_in[3];
    const float* W_mpl = (const float*)d_in[4];
    const float* b_mpl = (const float*)d_in[5];
    const float* W1 = (const float*)d_in[6];
    const float* b1 = (const float*)d_in[7];
    const float* W2 = (const float*)d_in[8];
    const float* b2 = (const float*)d_in[9];
    const float* W3 = (const float*)d_in[10];
    const float* b3 = (const float*)d_in[11];
    const float* W4 = (const float*)d_in[12];
    const float* b4 = (const float*)d_in[13];
    float* out = (float*)d_out;

    float* x_nodes = (float*)d_ws;                          // [N_NODES, 20]
    float* g       = x_nodes + (size_t)N_NODES * H1;        // [N_GRAPHS, 10]

    const size_t zero_bytes =
        ((size_t)N_NODES * H1 + (size_t)N_GRAPHS * H2) * sizeof(float);
    hipMemsetAsync(d_ws, 0, zero_bytes, stream);

    // 800000 edges / 32 per wave / 8 waves per block = 3125 blocks exactly.
    edge_mpl_kernel<<<N_EDGES / 32 / 8, 256, 0, stream>>>(
        edge_index, node_attr, edge_attr, W_mpl, b_mpl, x_nodes);

    node_mlp_kernel<<<(N_NODES + 255) / 256, 256, 0, stream>>>(
        x_nodes, batch, W1, b1, W2, b2, g);

    graph_mlp_kernel<<<(N_GRAPHS + 255) / 256, 256, 0, stream>>>(
        g, W3, b3, W4, b4, out);
}